// AttentionRGCN_68384469287431
// MI455X (gfx1250) — compile-verified
//
#include <hip/hip_runtime.h>

#define HDIM 256
#define EPSV 1e-5f

typedef __attribute__((ext_vector_type(2))) float v2f;
typedef __attribute__((ext_vector_type(8))) float v8f;

// ---------------------------------------------------------------------------
// gfx1250 async global->LDS copy (ASYNCcnt-tracked, no VGPR staging).
// ---------------------------------------------------------------------------
__device__ __forceinline__ void async_copy_b128(unsigned lds_off,
                                                unsigned long long gaddr) {
    asm volatile("global_load_async_to_lds_b128 %0, %1, off"
                 :: "v"(lds_off), "v"(gaddr)
                 : "memory");
}

__device__ __forceinline__ unsigned lds_off_of(const void* p) {
    return (unsigned)(uintptr_t)p;   // AS(3) pointer -> LDS byte offset
}

// ---------------------------------------------------------------------------
// C[M x 256] = A[M x K] @ W[K x 256] (+ bias), f32 via V_WMMA_F32_16X16X4_F32.
// Block: 256 threads (8 waves) -> 64 x 256 output tile.
// Wave w: rows m0 + 16*(w%4) .. +16, cols 128*(w/4) .. +128 (8 16x16 tiles).
//
// Double-buffered K-tiles of 16:
//   A tile (64x16, row-major, stride 20)  : async global->LDS, 1 instr/wave,
//                                           pipelined with s_wait_asynccnt 1.
//   W tile (16x256 -> transposed 256x16,  : global float4 loads issued one
//           stride 20, sWt[col][k])         tile ahead, scattered to LDS after
//                                           compute (overlapped with WMMA).
// Transposed W makes each B fragment (k, k+1 of one column) a single aligned
// ds_load_b64 into an even VGPR pair; stride 20 gives conflict-free fragment
// reads for both half-waves ({4m,4m+1} vs {4m+2,4m+3} bank pairs).
// ---------------------------------------------------------------------------
__global__ __launch_bounds__(256) void gemm_bias_kernel(
    const float* __restrict__ A, const float* __restrict__ W,
    const float* __restrict__ bias, float* __restrict__ C,
    int M, int K)
{
    __shared__ __align__(16) float sA [2][64 * 20];    // 2 x  5.0 KB
    __shared__ __align__(16) float sWt[2][256 * 20];   // 2 x 20.0 KB

    const int tid   = threadIdx.x;
    const int lane  = tid & 31;
    const int wave  = tid >> 5;
    const int l16   = lane & 15;
    const int hi    = lane >> 4;        // 0: lanes 0-15, 1: lanes 16-31
    const int khalf = hi << 1;          // A/B fragment K offset: 0 or 2

    const int m0      = blockIdx.x * 64;
    const int rowTile = wave & 3;           // 16-row strip within block tile
    const int colBase = (wave >> 2) * 128;  // 0 or 128

    // ---- A staging coords (async, 16B per thread) ----
    const int arow  = tid >> 2;               // 0..63
    const int acol  = (tid & 3) << 2;         // 0,4,8,12
    const int agrow = min(m0 + arow, M - 1);  // clamp; stores are row-guarded
    const unsigned sA_off[2] = {
        lds_off_of(&sA[0][arow * 20 + acol]),
        lds_off_of(&sA[1][arow * 20 + acol]) };

    auto issueA = [&](int k0, int buf) {
        async_copy_b128(sA_off[buf],
            (unsigned long long)(uintptr_t)(A + (size_t)agrow * K + k0 + acol));
    };

    // ---- W staging coords: thread = (row wk, col-quad wq) ----
    const int wk = tid >> 4;                  // 0..15 (K row)
    const int wq = tid & 15;                  // 0..15 (col quad)
    float4 wreg[4];

    auto loadW = [&](int k0) {                // dense: 16 lanes x 16B = 256B/instr
#pragma unroll
        for (int i = 0; i < 4; ++i)
            wreg[i] = *reinterpret_cast<const float4*>(
                W + (size_t)(k0 + wk) * HDIM + wq * 4 + 64 * i);
    };
    auto storeW = [&](int buf) {              // scatter to transposed layout
        float* p = sWt[buf];
#pragma unroll
        for (int i = 0; i < 4; ++i) {
            const int c = wq * 4 + 64 * i;
            p[(c + 0) * 20 + wk] = wreg[i].x;
            p[(c + 1) * 20 + wk] = wreg[i].y;
            p[(c + 2) * 20 + wk] = wreg[i].z;
            p[(c + 3) * 20 + wk] = wreg[i].w;
        }
    };

    v8f acc[8] = {};

    const int T = K >> 4;                   // K is a multiple of 16
    // prologue: tile 0
    issueA(0, 0);
    loadW(0);
    storeW(0);

    for (int t = 0; t < T; ++t) {
        const int cur = t & 1;
        const int nxt = cur ^ 1;
        if (t + 1 < T) {
            issueA((t + 1) << 4, nxt);      // async A prefetch
            loadW((t + 1) << 4);            // issue W loads (no wait yet)
            asm volatile("s_wait_asynccnt 0x1" ::: "memory");  // A(t) landed
        } else {
            asm volatile("s_wait_asynccnt 0x0" ::: "memory");
        }
        __syncthreads();                    // tile t visible to all waves

        const float* aB = sA[cur];
        const float* wB = sWt[cur];
#pragma unroll
        for (int kk = 0; kk < 4; ++kk) {
            const int ka = kk * 4 + khalf;
            // A fragment (16x4): lane = row l16, K = ka, ka+1 (one b64)
            const v2f afrag =
                *reinterpret_cast<const v2f*>(&aB[(rowTile * 16 + l16) * 20 + ka]);
#pragma unroll
            for (int j = 0; j < 8; ++j) {
                // B fragment (4x16): lane = col l16, K = ka, ka+1 (one b64)
                const int cb = colBase + j * 16 + l16;
                const v2f bfrag = *reinterpret_cast<const v2f*>(&wB[cb * 20 + ka]);
                acc[j] = __builtin_amdgcn_wmma_f32_16x16x4_f32(
                    false, afrag, false, bfrag, (short)0, acc[j], false, false);
            }
        }
        if (t + 1 < T) storeW(nxt);         // drain W regs after compute
        __syncthreads();                    // safe to overwrite / read next
    }

    // ---- store: C/D layout: VGPR v -> M = v (+8 for upper half-wave) ----
#pragma unroll
    for (int j = 0; j < 8; ++j) {
        const int col = colBase + j * 16 + l16;
        const float bv = bias ? bias[col] : 0.0f;
#pragma unroll
        for (int v = 0; v < 8; ++v) {
            const int row = m0 + rowTile * 16 + v + (hi << 3);
            if (row < M)
                C[(size_t)row * HDIM + col] = acc[j][v] + bv;
        }
    }
}

// ---------------------------------------------------------------------------
// Edge scatter: out[dst[e], :] += hw[src[e], :].
// 64 threads per edge, float4 gather + 4 hardware f32 atomic adds each.
// ---------------------------------------------------------------------------
__global__ __launch_bounds__(256) void scatter_add_kernel(
    const float* __restrict__ hw, const int* __restrict__ src,
    const int* __restrict__ dst, float* __restrict__ out, int E)
{
    const int e = blockIdx.x * 4 + (threadIdx.x >> 6);
    if (e >= E) return;
    const int c = (threadIdx.x & 63) << 2;
    const int s = src[e];
    const int d = dst[e];
    const float4 v = *reinterpret_cast<const float4*>(hw + (size_t)s * HDIM + c);
    float* q = out + (size_t)d * HDIM + c;
    unsafeAtomicAdd(q + 0, v.x);
    unsafeAtomicAdd(q + 1, v.y);
    unsafeAtomicAdd(q + 2, v.z);
    unsafeAtomicAdd(q + 3, v.w);
}

// ---------------------------------------------------------------------------
// LayerNorm (+optional residual), one wave32 per row (8 channels per lane).
// Writes result back into h.
// ---------------------------------------------------------------------------
__global__ __launch_bounds__(256) void layernorm_kernel(
    const float* __restrict__ outb, float* __restrict__ h,
    const float* __restrict__ g, const float* __restrict__ b,
    int N, int addResidual)
{
    const int lane = threadIdx.x & 31;
    const int row  = blockIdx.x * 8 + (threadIdx.x >> 5);
    if (row >= N) return;

    const float* xr = outb + (size_t)row * HDIM;
    float* hr = h + (size_t)row * HDIM;
    const int c0 = lane * 8;

    float x[8];
#pragma unroll
    for (int i = 0; i < 8; ++i) {
        x[i] = xr[c0 + i];
        if (addResidual) x[i] += hr[c0 + i];
    }

    float s = 0.f;
#pragma unroll
    for (int i = 0; i < 8; ++i) s += x[i];
#pragma unroll
    for (int m = 16; m >= 1; m >>= 1) s += __shfl_xor(s, m, 32);
    const float mu = s * (1.0f / 256.0f);

    float vs = 0.f;
#pragma unroll
    for (int i = 0; i < 8; ++i) { const float d = x[i] - mu; vs += d * d; }
#pragma unroll
    for (int m = 16; m >= 1; m >>= 1) vs += __shfl_xor(vs, m, 32);
    const float inv = rsqrtf(vs * (1.0f / 256.0f) + EPSV);

#pragma unroll
    for (int i = 0; i < 8; ++i)
        hr[c0 + i] = (x[i] - mu) * inv * g[c0 + i] + b[c0 + i];
}

// ---------------------------------------------------------------------------
extern "C" void kernel_launch(void* const* d_in, const int* in_sizes, int n_in,
                              void* d_out, int out_size, void* d_ws, size_t ws_size,
                              hipStream_t stream)
{
    const float* x_compound  = (const float*)d_in[0];
    const float* x_target    = (const float*)d_in[1];
    const float* x_disease   = (const float*)d_in[2];
    const int*   edge_index  = (const int*)d_in[3];
    const float* Wp_compound = (const float*)d_in[4];
    const float* bp_compound = (const float*)d_in[5];
    const float* Wp_target   = (const float*)d_in[6];
    const float* bp_target   = (const float*)d_in[7];
    const float* Wp_disease  = (const float*)d_in[8];
    const float* bp_disease  = (const float*)d_in[9];
    const float* rgcn_w      = (const float*)d_in[10]; // (L,R,H,H)
    const float* rgcn_root   = (const float*)d_in[11]; // (L,H,H)
    const float* rgcn_b      = (const float*)d_in[12]; // (L,H)
    const float* ln_g        = (const float*)d_in[13]; // (L,H)
    const float* ln_b        = (const float*)d_in[14]; // (L,H)
    const float* Wo          = (const float*)d_in[15]; // (H,H)
    const float* bo          = (const float*)d_in[16]; // (H)

    const int N_C = 50000, N_T = 30000, N_D = 20000;
    const int N = N_C + N_T + N_D;
    const int R = 4, E_R = 400000, L = 2;

    float* h  = (float*)d_ws;                 // N x 256
    float* ob = h  + (size_t)N * HDIM;        // N x 256 (pre-LN accumulator)
    float* hw = ob + (size_t)N * HDIM;        // N x 256 (per-relation h @ W)

    const dim3 blk(256);
    const dim3 gN((N + 63) / 64);

    // Projections into h
    gemm_bias_kernel<<<dim3((N_C + 63) / 64), blk, 0, stream>>>(
        x_compound, Wp_compound, bp_compound, h, N_C, 256);
    gemm_bias_kernel<<<dim3((N_T + 63) / 64), blk, 0, stream>>>(
        x_target, Wp_target, bp_target, h + (size_t)N_C * HDIM, N_T, 512);
    gemm_bias_kernel<<<dim3((N_D + 63) / 64), blk, 0, stream>>>(
        x_disease, Wp_disease, bp_disease, h + (size_t)(N_C + N_T) * HDIM, N_D, 128);

    for (int l = 0; l < L; ++l) {
        // out = h @ root[l] + b[l]   (GEMM writes every element: no init needed)
        gemm_bias_kernel<<<gN, blk, 0, stream>>>(
            h, rgcn_root + (size_t)l * HDIM * HDIM, rgcn_b + (size_t)l * HDIM,
            ob, N, 256);
        for (int r = 0; r < R; ++r) {
            // hw = h @ rgcn_w[l][r]
            gemm_bias_kernel<<<gN, blk, 0, stream>>>(
                h, rgcn_w + ((size_t)l * R + r) * HDIM * HDIM, nullptr, hw, N, 256);
            const int* src = edge_index + (size_t)r * E_R;
            const int* dst = edge_index + (size_t)R * E_R + (size_t)r * E_R;
            scatter_add_kernel<<<dim3(E_R / 4), blk, 0, stream>>>(hw, src, dst, ob, E_R);
        }
        // h = LayerNorm(out (+h if l>0))
        layernorm_kernel<<<dim3((N + 7) / 8), blk, 0, stream>>>(
            ob, h, ln_g + (size_t)l * HDIM, ln_b + (size_t)l * HDIM, N, l > 0);
    }

    // y = h @ Wo + bo  ->  d_out (flat concatenation of the three slices == full y)
    gemm_bias_kernel<<<gN, blk, 0, stream>>>(h, Wo, bo, (float*)d_out, N, 256);
}